// BiGRU_5686536700109
// MI455X (gfx1250) — compile-verified
//
#include <hip/hip_runtime.h>

// BiGRU on MI455X (gfx1250): fp32 WMMA (V_WMMA_F32_16X16X4_F32) everywhere.
// Phase 1: 6 input-projection GEMMs -> d_ws (if ws big enough).
// Phase 2: 2048 sequential step kernels, h carried through d_out slices.
// Phase 3: copy out[-1] -> h_last tail of d_out.

typedef float v2f __attribute__((ext_vector_type(2)));
typedef float v8f __attribute__((ext_vector_type(8)));

#define S_ 2048
#define B_ 32
#define I_ 512
#define H_ 512
#define R_ (S_ * B_)          // 65536 flattened rows of x
#define OUTROW_ (B_ * 2 * H_) // 32768 floats per timestep slice of out

static __device__ __forceinline__ v8f wmma_f32(v2f a, v2f b, v8f c) {
  // D = A(16x4 f32) * B(4x16 f32) + C(16x16 f32)
  return __builtin_amdgcn_wmma_f32_16x16x4_f32(false, a, false, b, (short)0, c,
                                               false, false);
}

static __device__ __forceinline__ float sigmoid_f(float x) {
  return 1.f / (1.f + __expf(-x)); // saturates cleanly at 0/1
}
static __device__ __forceinline__ float tanh_f(float x) {
  float e = __expf(2.f * x);       // inf-safe form: 1 - 2/(e+1)
  return 1.f - 2.f / (e + 1.f);
}

struct GruParams {
  const float* Wh[2][3]; // [dir][r,z,n]  (H x H, row-major [k][n])
  const float* bh[2][3];
  const float* Wi[2][3]; // used by fused variant
  const float* bi[2][3];
  const float* x;        // [S,B,I]
  const float* ws;       // xproj: [dir*3+gate][S*B][H]
  float* out;            // [S,B,2H] (+ tail [B,2H])
};

// ---------------------------------------------------------------------------
// Phase 1: xproj[g] = x @ Wi[g] + bi[g] for g in {f_r,f_z,f_n,b_r,b_z,b_n}
// One wave per 16x16 output tile position, computing all 6 projections so the
// A fragment (x) is loaded once and reused 6x.
// grid = (R/16) * (32 ntiles / 4 waves) = 4096*8 blocks, 128 thr.
// ---------------------------------------------------------------------------
__global__ void __launch_bounds__(128) gru_xproj(GruParams p) {
  const int lane  = threadIdx.x & 31;
  const int wave  = threadIdx.x >> 5;
  const int mtile = blockIdx.x >> 3;
  const int ntile = ((blockIdx.x & 7) << 2) | wave;
  const int m0 = mtile * 16, n0 = ntile * 16;
  const int mn = lane & 15;          // A: row m ; B: col n
  const int kb = (lane >> 4) << 1;   // lanes 16-31 hold K+2

  v8f acc[6] = {};
  const float* xrow = p.x + (size_t)(m0 + mn) * I_;
  for (int k0 = 0; k0 < I_; k0 += 4) {
    v2f a;
    a.x = xrow[k0 + kb];
    a.y = xrow[k0 + kb + 1];
    const int w0 = (k0 + kb) * H_ + n0 + mn;
    const int w1 = w0 + H_;
#pragma unroll
    for (int d = 0; d < 2; ++d)
#pragma unroll
      for (int g = 0; g < 3; ++g) {
        v2f b;
        b.x = p.Wi[d][g][w0];
        b.y = p.Wi[d][g][w1];
        acc[d * 3 + g] = wmma_f32(a, b, acc[d * 3 + g]);
      }
  }
  const int n   = n0 + (lane & 15);
  const int mhi = (lane >> 4) << 3;  // C/D: lanes 16-31 hold rows M+8
#pragma unroll
  for (int d = 0; d < 2; ++d)
#pragma unroll
    for (int g = 0; g < 3; ++g) {
      const float bias = p.bi[d][g][n];
      float* o = const_cast<float*>(p.ws) +
                 ((size_t)(d * 3 + g) * R_ + (size_t)(m0 + mhi)) * H_ + n;
#pragma unroll
      for (int v = 0; v < 8; ++v) o[(size_t)v * H_] = acc[d * 3 + g][v] + bias;
    }
}

// ---------------------------------------------------------------------------
// Phase 2: one GRU step. grid = 2 dirs * 2 mtiles * 8 ngroups = 32 blocks,
// 4 waves/block, one 16x16 h_new tile per wave, 3 (or 6) gate GEMMs fused.
// h_{t-1} read from out[t-1] slice; weights stay resident in the 192MB L2.
// ---------------------------------------------------------------------------
template <bool FUSED>
__global__ void __launch_bounds__(128) gru_step(GruParams p, int t) {
  const int lane  = threadIdx.x & 31;
  const int wave  = threadIdx.x >> 5;
  const int dir   = blockIdx.x >> 4;
  const int mtile = (blockIdx.x >> 3) & 1;
  const int ntile = ((blockIdx.x & 7) << 2) | wave;
  const int m0 = mtile * 16, n0 = ntile * 16;
  const int mn = lane & 15;
  const int kb = (lane >> 4) << 1;

  const float* Whr = p.Wh[dir][0];
  const float* Whz = p.Wh[dir][1];
  const float* Whn = p.Wh[dir][2];

  v8f ar = {}, az = {}, anh = {}, anx = {};

  if (t > 0) { // recurrent GEMMs: h_{t-1} @ Wh{r,z,n}
    const float* hrow = p.out + (size_t)(t - 1) * OUTROW_ + dir * H_ +
                        (size_t)(m0 + mn) * (2 * H_);
    for (int k0 = 0; k0 < H_; k0 += 4) {
      v2f a;
      a.x = hrow[k0 + kb];
      a.y = hrow[k0 + kb + 1];
      const int w0 = (k0 + kb) * H_ + n0 + mn;
      const int w1 = w0 + H_;
      v2f b;
      b.x = Whr[w0]; b.y = Whr[w1]; ar  = wmma_f32(a, b, ar);
      b.x = Whz[w0]; b.y = Whz[w1]; az  = wmma_f32(a, b, az);
      b.x = Whn[w0]; b.y = Whn[w1]; anh = wmma_f32(a, b, anh);
    }
  }
  if (FUSED) { // fold x_t @ Wi{r,z,n} in (n-gate kept separate: outside r*(.))
    const float* Wir = p.Wi[dir][0];
    const float* Wiz = p.Wi[dir][1];
    const float* Win = p.Wi[dir][2];
    const float* xrow = p.x + ((size_t)t * B_ + (size_t)(m0 + mn)) * I_;
    for (int k0 = 0; k0 < I_; k0 += 4) {
      v2f a;
      a.x = xrow[k0 + kb];
      a.y = xrow[k0 + kb + 1];
      const int w0 = (k0 + kb) * H_ + n0 + mn;
      const int w1 = w0 + H_;
      v2f b;
      b.x = Wir[w0]; b.y = Wir[w1]; ar  = wmma_f32(a, b, ar);
      b.x = Wiz[w0]; b.y = Wiz[w1]; az  = wmma_f32(a, b, az);
      b.x = Win[w0]; b.y = Win[w1]; anx = wmma_f32(a, b, anx);
    }
  }

  // gate math + h_new
  const int n   = n0 + (lane & 15);
  const int mhi = (lane >> 4) << 3;
  const float vbhr = p.bh[dir][0][n];
  const float vbhz = p.bh[dir][1][n];
  const float vbhn = p.bh[dir][2][n];
  float vbir = 0.f, vbiz = 0.f, vbin = 0.f;
  if (FUSED) {
    vbir = p.bi[dir][0][n];
    vbiz = p.bi[dir][1][n];
    vbin = p.bi[dir][2][n];
  }
  const float* hp = (t > 0)
      ? p.out + (size_t)(t - 1) * OUTROW_ + dir * H_ : nullptr;
  float* op = p.out + (size_t)t * OUTROW_ + dir * H_;
  const float* wsr = p.ws + ((size_t)(dir * 3 + 0) * R_ + (size_t)t * B_) * H_;
  const float* wsz = p.ws + ((size_t)(dir * 3 + 1) * R_ + (size_t)t * B_) * H_;
  const float* wsn = p.ws + ((size_t)(dir * 3 + 2) * R_ + (size_t)t * B_) * H_;

#pragma unroll
  for (int v = 0; v < 8; ++v) {
    const int brow = m0 + mhi + v; // batch index 0..31
    const float xr_t = FUSED ? vbir : wsr[(size_t)brow * H_ + n];
    const float xz_t = FUSED ? vbiz : wsz[(size_t)brow * H_ + n];
    const float xn_t = FUSED ? (anx[v] + vbin) : wsn[(size_t)brow * H_ + n];
    const float r = sigmoid_f(xr_t + ar[v] + vbhr);
    const float z = sigmoid_f(xz_t + az[v] + vbhz);
    const float nn = tanh_f(xn_t + r * (anh[v] + vbhn));
    const float hprev = hp ? hp[(size_t)brow * (2 * H_) + n] : 0.f;
    op[(size_t)brow * (2 * H_) + n] = (1.f - z) * nn + z * hprev;
  }
}

// ---------------------------------------------------------------------------
// Phase 3: h_last = out[S-1]
// ---------------------------------------------------------------------------
__global__ void gru_tail(float* out) {
  const int i = blockIdx.x * blockDim.x + threadIdx.x;
  if (i < OUTROW_)
    out[(size_t)S_ * OUTROW_ + i] = out[(size_t)(S_ - 1) * OUTROW_ + i];
}

extern "C" void kernel_launch(void* const* d_in, const int* in_sizes, int n_in,
                              void* d_out, int out_size, void* d_ws,
                              size_t ws_size, hipStream_t stream) {
  (void)in_sizes; (void)n_in; (void)out_size;
  // input order: x, then f_{Wir,Whr,Wiz,Whz,Win,Whn,bir,bhr,biz,bhz,bin,bhn},
  // then b_{...} in the same order.
  GruParams p{};
  p.x   = (const float*)d_in[0];
  p.ws  = (const float*)d_ws;
  p.out = (float*)d_out;
  for (int d = 0; d < 2; ++d) {
    const int base = d ? 13 : 1;
    auto G = [&](int i) { return (const float*)d_in[base + i]; };
    p.Wi[d][0] = G(0);  p.Wh[d][0] = G(1);
    p.Wi[d][1] = G(2);  p.Wh[d][1] = G(3);
    p.Wi[d][2] = G(4);  p.Wh[d][2] = G(5);
    p.bi[d][0] = G(6);  p.bh[d][0] = G(7);
    p.bi[d][1] = G(8);  p.bh[d][1] = G(9);
    p.bi[d][2] = G(10); p.bh[d][2] = G(11);
  }

  const size_t need = (size_t)6 * R_ * H_ * sizeof(float); // 768 MB xproj
  if (ws_size >= need) {
    gru_xproj<<<(R_ / 16) * 8, 128, 0, stream>>>(p);
    for (int t = 0; t < S_; ++t)
      gru_step<false><<<32, 128, 0, stream>>>(p, t);
  } else {
    for (int t = 0; t < S_; ++t)
      gru_step<true><<<32, 128, 0, stream>>>(p, t);
  }
  gru_tail<<<(OUTROW_ + 255) / 256, 256, 0, stream>>>((float*)d_out);
}